// Fusion_46557445489053
// MI455X (gfx1250) — compile-verified
//
#include <hip/hip_runtime.h>
#include <hip/hip_bf16.h>
#include <stdint.h>

// Problem constants (match reference)
#define Hc   512
#define Wc   512
#define HWc  (Hc * Wc)
#define Bc   4
#define Cc   128
#define Kc   512

// NMS tiling: output tile 32x32, halo = 3 * 5 chained maxpools = 15
#define TILE 32
#define HALO 15
#define R    62          // TILE + 2*HALO
#define RS   64          // padded row stride
#define NMS_NT 256

#define EQCAP 2048       // tie-bucket capacity for top-k

// ---------- helpers ----------

__device__ __forceinline__ unsigned f2k(float f) {
  // order-preserving float -> uint (ascending)
  unsigned u = __float_as_uint(f);
  return (u & 0x80000000u) ? ~u : (u | 0x80000000u);
}

__device__ __forceinline__ float rowmax7f(const float* r, int x) {
  float m = r[x - 3];
  m = fmaxf(m, r[x - 2]); m = fmaxf(m, r[x - 1]); m = fmaxf(m, r[x]);
  m = fmaxf(m, r[x + 1]); m = fmaxf(m, r[x + 2]); m = fmaxf(m, r[x + 3]);
  return m;
}

__device__ __forceinline__ float colmax7f(const float* a, int y, int x) {
  float m =         a[(y - 3) * RS + x];
  m = fmaxf(m, a[(y - 2) * RS + x]);
  m = fmaxf(m, a[(y - 1) * RS + x]);
  m = fmaxf(m, a[(y    ) * RS + x]);
  m = fmaxf(m, a[(y + 1) * RS + x]);
  m = fmaxf(m, a[(y + 2) * RS + x]);
  m = fmaxf(m, a[(y + 3) * RS + x]);
  return m;
}

// ---------- kernel 1: fused NMS (5 maxpools) + mask -> flat masked scores ----------

__global__ __launch_bounds__(NMS_NT)
void nms_mask_kernel(const float* __restrict__ score, float* __restrict__ flat) {
  __shared__ float S[R * RS];          // raw scores tile (with -inf halo padding)
  __shared__ float A[R * RS];          // row-pass temp
  __shared__ float Cm[R * RS];         // suppressed scores
  __shared__ unsigned char M[R * RS];  // running max_mask
  __shared__ unsigned char P[R * RS];  // supp_mask

  const int tid = threadIdx.x;
  const int b   = blockIdx.z;
  const int gx0 = (int)blockIdx.x * TILE - HALO;
  const int gy0 = (int)blockIdx.y * TILE - HALO;
  // Uniform per-block base address -> lives in SGPRs for the async SADDR form.
  const unsigned long long base =
      (unsigned long long)(size_t)(score + (size_t)b * HWc);

  // Pre-fill with -inf (reduce_window pads with -inf outside the image).
  for (int i = tid; i < R * RS; i += NMS_NT) S[i] = -__builtin_inff();
  __syncthreads();
  // Ensure DS init stores are complete before async LDS writes can land
  // (async ops are unordered vs DS ops).
  asm volatile("s_wait_dscnt 0" ::: "memory");

  // CDNA5 async copy: global -> LDS, one dword per enabled lane.
  for (int i = tid; i < R * R; i += NMS_NT) {
    int y = i / R, x = i - y * R;
    int gy = gy0 + y, gx = gx0 + x;
    if ((unsigned)gy < (unsigned)Hc && (unsigned)gx < (unsigned)Wc) {
      unsigned lds = (unsigned)(size_t)&S[y * RS + x];   // low 32 bits = LDS offset
      unsigned off = (unsigned)((gy * Wc + gx) * 4);
      asm volatile("global_load_async_to_lds_b32 %0, %1, %2"
                   :: "v"(lds), "v"(off), "s"(base) : "memory");
    }
  }
  asm volatile("s_wait_asynccnt 0" ::: "memory");
  __syncthreads();

  // Stage 0: max_mask = (S == maxpool(S)), valid on [3, R-3)^2
  {
    const int y0 = 0, y1 = R, x0 = 3, x1 = R - 3, wdt = x1 - x0;
    const int n = (y1 - y0) * wdt;
    for (int i = tid; i < n; i += NMS_NT) {
      int y = y0 + i / wdt, x = x0 + i % wdt;
      A[y * RS + x] = rowmax7f(&S[y * RS], x);
    }
  }
  __syncthreads();
  {
    const int y0 = 3, y1 = R - 3, x0 = 3, x1 = R - 3, wdt = x1 - x0;
    const int n = (y1 - y0) * wdt;
    for (int i = tid; i < n; i += NMS_NT) {
      int y = y0 + i / wdt, x = x0 + i % wdt;
      float m = colmax7f(A, y, x);
      M[y * RS + x] = (S[y * RS + x] == m) ? 1 : 0;
    }
  }
  __syncthreads();

  // Two suppression iterations; regions shrink by 6 each iteration.
#pragma unroll
  for (int it = 0; it < 2; ++it) {
    const int m0 = 3 + 6 * it;   // mask region border
    const int s0 = 6 + 6 * it;   // supp / supp_scores region border
    const int u0 = 9 + 6 * it;   // update region border (it==1 -> HALO)

    // row-max of mask bytes -> A (float)
    {
      const int y0 = m0, y1 = R - m0, x0 = s0, x1 = R - s0, wdt = x1 - x0;
      const int n = (y1 - y0) * wdt;
      for (int i = tid; i < n; i += NMS_NT) {
        int y = y0 + i / wdt, x = x0 + i % wdt;
        const unsigned char* r = &M[y * RS];
        int m = r[x - 3];
        m = max(m, (int)r[x - 2]); m = max(m, (int)r[x - 1]); m = max(m, (int)r[x]);
        m = max(m, (int)r[x + 1]); m = max(m, (int)r[x + 2]); m = max(m, (int)r[x + 3]);
        A[y * RS + x] = (float)m;
      }
    }
    __syncthreads();
    // col-max -> supp mask P; Cm = supp ? 0 : S
    {
      const int y0 = s0, y1 = R - s0, x0 = s0, x1 = R - s0, wdt = x1 - x0;
      const int n = (y1 - y0) * wdt;
      for (int i = tid; i < n; i += NMS_NT) {
        int y = y0 + i / wdt, x = x0 + i % wdt;
        float m = colmax7f(A, y, x);
        int supp = (m > 0.0f) ? 1 : 0;
        P[y * RS + x]  = (unsigned char)supp;
        Cm[y * RS + x] = supp ? 0.0f : S[y * RS + x];
      }
    }
    __syncthreads();
    // row-max of suppressed scores -> A
    {
      const int y0 = s0, y1 = R - s0, x0 = u0, x1 = R - u0, wdt = x1 - x0;
      const int n = (y1 - y0) * wdt;
      for (int i = tid; i < n; i += NMS_NT) {
        int y = y0 + i / wdt, x = x0 + i % wdt;
        A[y * RS + x] = rowmax7f(&Cm[y * RS], x);
      }
    }
    __syncthreads();

    if (it == 0) {
      // M |= (Cm == maxpool(Cm)) & ~P   over [u0, R-u0)^2
      const int y0 = u0, y1 = R - u0, x0 = u0, x1 = R - u0, wdt = x1 - x0;
      const int n = (y1 - y0) * wdt;
      for (int i = tid; i < n; i += NMS_NT) {
        int y = y0 + i / wdt, x = x0 + i % wdt;
        float m = colmax7f(A, y, x);
        bool nm = (Cm[y * RS + x] == m);
        bool sp = (P[y * RS + x] != 0);
        bool mm = (M[y * RS + x] != 0) || (nm && !sp);
        M[y * RS + x] = mm ? 1 : 0;
      }
      __syncthreads();
    } else {
      // Final: flat = (max_mask && s>0) ? s : -1 over the 32x32 output tile.
      const int y0 = HALO, y1 = R - HALO, x0 = HALO, x1 = R - HALO, wdt = x1 - x0;
      const int n = (y1 - y0) * wdt;
      for (int i = tid; i < n; i += NMS_NT) {
        int y = y0 + i / wdt, x = x0 + i % wdt;
        float m = colmax7f(A, y, x);
        bool nm = (Cm[y * RS + x] == m);
        bool sp = (P[y * RS + x] != 0);
        bool mm = (M[y * RS + x] != 0) || (nm && !sp);
        float s = S[y * RS + x];
        float o = (mm && s > 0.0f) ? s : -1.0f;
        int gy = gy0 + y, gx = gx0 + x;
        flat[(size_t)b * HWc + gy * Wc + gx] = o;
      }
    }
  }
}

// ---------- kernel 2: per-batch top-K via radix select + bitonic sort ----------

__global__ __launch_bounds__(1024)
void topk_kernel(const float* __restrict__ flat,
                 const float* __restrict__ points,
                 float* __restrict__ out_kpts,   // (B,K,4)
                 int*   __restrict__ out_pix,    // (B,K,2)
                 int*   __restrict__ idx_ws) {   // (B,K)
  const int b   = blockIdx.x;
  const int tid = threadIdx.x;
  const float* p = flat + (size_t)b * HWc;

  __shared__ unsigned hist[256];
  __shared__ unsigned long long cand[Kc];  // key<<32 | ~idx  (sort desc => idx asc ties)
  __shared__ int eq[EQCAP];
  __shared__ unsigned s_bucket, s_acc, s_cntG, s_cntE;

  // 4-pass MSB-first radix select for the K-th largest key.
  unsigned prefix = 0;
  int need = Kc;
  for (int pass = 0; pass < 4; ++pass) {
    const int shift = 24 - pass * 8;
    if (tid < 256) hist[tid] = 0;
    __syncthreads();
    for (int i = tid; i < HWc; i += 1024) {
      unsigned key = f2k(p[i]);
      bool ok = (pass == 0) || ((key >> (shift + 8)) == prefix);
      if (ok) atomicAdd(&hist[(key >> shift) & 255u], 1u);
    }
    __syncthreads();
    if (tid == 0) {
      unsigned acc = 0;
      int bkt = 255;
      for (; bkt > 0; --bkt) {
        unsigned c = hist[bkt];
        if (acc + c >= (unsigned)need) break;
        acc += c;
      }
      s_bucket = (unsigned)bkt;
      s_acc = acc;
    }
    __syncthreads();
    prefix = (prefix << 8) | s_bucket;
    need -= (int)s_acc;
  }
  const unsigned thresh = prefix;   // K-th largest key; take `need` of the == bucket

  // Compaction: strictly-greater go straight into cand; ties collected separately.
  if (tid == 0) { s_cntG = 0; s_cntE = 0; }
  __syncthreads();
  for (int i = tid; i < HWc; i += 1024) {
    unsigned key = f2k(p[i]);
    if (key > thresh) {
      unsigned pos = atomicAdd(&s_cntG, 1u);
      if (pos < (unsigned)Kc)
        cand[pos] = ((unsigned long long)key << 32) | (unsigned)(~i);
    } else if (key == thresh) {
      unsigned pos = atomicAdd(&s_cntE, 1u);
      if (pos < (unsigned)EQCAP) eq[pos] = i;
    }
  }
  __syncthreads();
  const unsigned G = s_cntG;
  const unsigned e = (s_cntE < (unsigned)EQCAP) ? s_cntE : (unsigned)EQCAP;

  // Sort tie indices ascending (JAX top_k tie-break: lower index first).
  for (int i = tid; i < EQCAP; i += 1024) if (i >= (int)e) eq[i] = 0x7FFFFFFF;
  __syncthreads();
  for (int k = 2; k <= EQCAP; k <<= 1) {
    for (int j = k >> 1; j > 0; j >>= 1) {
      for (int i = tid; i < EQCAP; i += 1024) {
        int ix = i ^ j;
        if (ix > i) {
          int a = eq[i], c = eq[ix];
          bool up = ((i & k) == 0);
          if (up ? (a > c) : (a < c)) { eq[i] = c; eq[ix] = a; }
        }
      }
      __syncthreads();
    }
  }

  // Append `take` smallest tie indices at the threshold value.
  int take = Kc - (int)G; if (take < 0) take = 0;
  for (int t = tid; t < take; t += 1024) {
    int ii = (t < (int)e) ? eq[t] : 0;
    if ((int)G + t < Kc)
      cand[G + t] = ((unsigned long long)thresh << 32) | (unsigned)(~ii);
  }
  __syncthreads();

  // Bitonic sort cand[0..K) descending (=> value desc, index asc on ties).
  for (int k = 2; k <= Kc; k <<= 1) {
    for (int j = k >> 1; j > 0; j >>= 1) {
      if (tid < Kc) {
        int i = tid, ix = i ^ j;
        if (ix > i) {
          unsigned long long a = cand[i], c = cand[ix];
          bool up = ((i & k) == 0);
          if (up ? (a < c) : (a > c)) { cand[i] = c; cand[ix] = a; }
        }
      }
      __syncthreads();
    }
  }

  // Emit kpts, pixel coords, and the index list for the feature gather.
  const float* ptsb = points + (size_t)b * 4 * HWc;
  for (int t = tid; t < Kc; t += 1024) {
    unsigned long long c = cand[t];
    int idx = (int)(~(unsigned)c) & (HWc - 1);
    idx_ws[b * Kc + t] = idx;
    int u = idx >> 9;        // idx / W
    int v = idx & (Wc - 1);  // idx % W
    size_t ko = ((size_t)b * Kc + t) * 4;
    out_kpts[ko + 0] = ptsb[idx];           // points ch 0
    out_kpts[ko + 1] = ptsb[HWc + idx];     // points ch 1
    out_kpts[ko + 2] = 0.0f;                // ch 2 forced to 0
    out_kpts[ko + 3] = 1.0f;                // ch 3 forced to 1
    size_t po = ((size_t)b * Kc + t) * 2;
    out_pix[po + 0] = u;
    out_pix[po + 1] = v;
  }
}

// ---------- kernel 3: feature gather (B,C,K) ----------

__global__ __launch_bounds__(Kc)
void gather_feat_kernel(const float* __restrict__ feat,
                        const int* __restrict__ idx_ws,
                        float* __restrict__ out_feas) {
  const int c = blockIdx.x;       // 0..C-1
  const int b = blockIdx.y;       // 0..B-1
  const int j = threadIdx.x;      // 0..K-1  (coalesced writes)
  const int idx = idx_ws[b * Kc + j];
  out_feas[((size_t)b * Cc + c) * Kc + j] =
      feat[((size_t)b * Cc + c) * HWc + idx];
}

// ---------- launch ----------

extern "C" void kernel_launch(void* const* d_in, const int* in_sizes, int n_in,
                              void* d_out, int out_size, void* d_ws, size_t ws_size,
                              hipStream_t stream) {
  (void)in_sizes; (void)n_in; (void)out_size; (void)ws_size;
  const float* score  = (const float*)d_in[0];   // (B,1,H,W)
  const float* feat   = (const float*)d_in[1];   // (B,C,H,W)
  const float* points = (const float*)d_in[2];   // (B,4,H,W)

  float* out = (float*)d_out;
  float* out_kpts = out;                                        // B*K*4   = 8192
  float* out_feas = out + (size_t)Bc * Kc * 4;                  // B*C*K   = 262144
  int*   out_pix  = (int*)(out + (size_t)Bc * Kc * 4 + (size_t)Bc * Cc * Kc);

  float* flat   = (float*)d_ws;                                 // B*HW floats
  int*   idx_ws = (int*)((char*)d_ws + sizeof(float) * (size_t)Bc * HWc);

  nms_mask_kernel<<<dim3(Wc / TILE, Hc / TILE, Bc), NMS_NT, 0, stream>>>(score, flat);
  topk_kernel<<<dim3(Bc), 1024, 0, stream>>>(flat, points, out_kpts, out_pix, idx_ws);
  gather_feat_kernel<<<dim3(Cc, Bc), Kc, 0, stream>>>(feat, idx_ws, out_feas);
}